// RegionProposalNetwork_20950850469892
// MI455X (gfx1250) — compile-verified
//
#include <hip/hip_runtime.h>
#include <hip/hip_bf16.h>
#include <math.h>

typedef __attribute__((ext_vector_type(2))) float v2f;
typedef __attribute__((ext_vector_type(8))) float v8f;

#define NIMG 4
#define CIN 256
#define HF 128
#define WF 128
#define A_ANCH 3
#define NPROP (A_ANCH * HF * WF)   // 49152
#define NPAD 65536
#define PRE_NMS 2000
#define POST_NMS 300
#define OBJ_THR 0.3f
#define IOU_THR 0.7f
#define TILE_W 64                   // pixels per block (1 row x 64)
#define CCHUNK 32                   // input channels staged per LDS chunk
#define CIPAD 34                    // ci padded stride in LDS

// ---------------------------------------------------------------------------
// K0: repack conv weights:  wt2[(kh*3+kw)][ci/2][co][2]  (pairs of adjacent ci)
// ---------------------------------------------------------------------------
__global__ __launch_bounds__(256) void repack_w_kernel(const float* __restrict__ conv_w,
                                                       float* __restrict__ wt2) {
    int t = blockIdx.x * 256 + threadIdx.x;          // 9*128*256*2 = 589824 exact
    int par = t & 1;
    int co  = (t >> 1) & 255;
    int q   = t >> 9;                                 // kk*128 + ci2
    int ci2 = q & 127;
    int kk  = q >> 7;
    int ci  = ci2 * 2 + par;
    int kh  = kk / 3, kw = kk % 3;
    wt2[t] = conv_w[((co * CIN + ci) * 3 + kh) * 3 + kw];
}

// ---------------------------------------------------------------------------
// K1: fused 3x3 conv (implicit GEMM via v_wmma_f32_16x16x4_f32) + head convs
//   grid: NIMG * HF * (WF/TILE_W) = 1024 blocks of 256 threads (8 waves)
//   per wave: M-tiles = 2 (32 output channels), N-tiles = 4 (64 pixels)
// ---------------------------------------------------------------------------
__global__ __launch_bounds__(256) void conv_fused_kernel(
    const float* __restrict__ feat, const float* __restrict__ wt2,
    const float* __restrict__ conv_b,
    const float* __restrict__ det_w, const float* __restrict__ det_b,
    const float* __restrict__ reg_w, const float* __restrict__ reg_b,
    float* __restrict__ scores, float* __restrict__ deltas) {

    __shared__ float lds_in[3 * 66 * CIPAD];  // [r][col][ci]  (26,928 B)
    __shared__ float x_tile[256 * 33];        // [co][pix32]   (33,792 B)
    __shared__ float hb[16];

    int bid = blockIdx.x;
    int n    = bid >> 8;                    // 256 blocks per image
    int rem  = bid & 255;
    int row  = rem >> 1;
    int col0 = (rem & 1) * TILE_W;

    int tid  = threadIdx.x;
    int wave = tid >> 5;
    int lane = tid & 31;
    int hl   = lane >> 4;    // half-wave selector (K pair / M row group)
    int lm   = lane & 15;

    v8f acc[2][4] = {};

    const float* featn = feat + (size_t)n * CIN * HF * WF;
    const v2f*   wp    = (const v2f*)wt2;

    for (int cib = 0; cib < CIN; cib += CCHUNK) {
        __syncthreads();
        // stage 32-channel input chunk with halo into LDS (zero-pad OOB)
        for (int e = tid; e < 3 * 66 * CCHUNK; e += 256) {
            int col = e % 66;
            int r   = (e / 66) % 3;
            int cl  = e / 198;
            int grow = row + r - 1;
            int gcol = col0 + col - 1;
            float v = 0.0f;
            if (grow >= 0 && grow < HF && gcol >= 0 && gcol < WF)
                v = featn[((size_t)(cib + cl) * HF + grow) * WF + gcol];
            lds_in[(r * 66 + col) * CIPAD + cl] = v;
        }
        __syncthreads();

        for (int kh = 0; kh < 3; kh++) {
            for (int kw = 0; kw < 3; kw++) {
                int kk = kh * 3 + kw;
                #pragma unroll 4
                for (int c = 0; c < CCHUNK; c += 4) {
                    int ci2 = ((cib + c) >> 1) + hl;
                    int co0 = wave * 32 + lm;
                    // A fragments (16x4 f32): lane<16 holds K0,K1; lane>=16 K2,K3
                    v2f a0 = wp[(kk * 128 + ci2) * 256 + co0];
                    v2f a1 = wp[(kk * 128 + ci2) * 256 + co0 + 16];
                    // B fragments (4x16 f32) from LDS: N = lm, K pair by half
                    int boff = c + 2 * hl;
                    v2f b[4];
                    #pragma unroll
                    for (int nt = 0; nt < 4; nt++)
                        b[nt] = *(const v2f*)&lds_in[(kh * 66 + nt * 16 + lm + kw) * CIPAD + boff];
                    #pragma unroll
                    for (int nt = 0; nt < 4; nt++) {
                        acc[0][nt] = __builtin_amdgcn_wmma_f32_16x16x4_f32(false, a0, false, b[nt], (short)0, acc[0][nt], false, false);
                        acc[1][nt] = __builtin_amdgcn_wmma_f32_16x16x4_f32(false, a1, false, b[nt], (short)0, acc[1][nt], false, false);
                    }
                }
            }
        }
    }

    __syncthreads();   // all WMMA LDS reads complete; lds_in is free
    // stage head weights (det 3x256 then reg 12x256) into lds_in (6732 floats)
    for (int e = tid; e < 15 * 256; e += 256)
        lds_in[e] = (e < 768) ? det_w[e] : reg_w[e - 768];
    if (tid < 15) hb[tid] = (tid < 3) ? det_b[tid] : reg_b[tid - 3];

    // bias values for this lane's output-channel rows
    int m0  = hl * 8;
    float biasA[8], biasB[8];
    #pragma unroll
    for (int r = 0; r < 8; r++) {
        biasA[r] = conv_b[wave * 32 + m0 + r];
        biasB[r] = conv_b[wave * 32 + 16 + m0 + r];
    }

    // two 32-pixel phases reusing x_tile
    for (int ph = 0; ph < 2; ph++) {
        if (ph) __syncthreads();   // heads of phase 0 done reading x_tile
        // scatter D (16x16 f32 layout: VGPR r -> M = 8*half + r), add bias
        #pragma unroll
        for (int r = 0; r < 8; r++) {
            int coA = wave * 32 + m0 + r;
            int coB = wave * 32 + 16 + m0 + r;
            x_tile[coA * 33 + lm]      = acc[0][ph * 2 + 0][r] + biasA[r];
            x_tile[coA * 33 + 16 + lm] = acc[0][ph * 2 + 1][r] + biasA[r];
            x_tile[coB * 33 + lm]      = acc[1][ph * 2 + 0][r] + biasB[r];
            x_tile[coB * 33 + 16 + lm] = acc[1][ph * 2 + 1][r] + biasB[r];
        }
        __syncthreads();

        // 15 outputs x 32 pixels of 256-length dot products
        for (int t = tid; t < 15 * 32; t += 256) {
            int oi  = t >> 5;
            int pix = t & 31;
            float sum = hb[oi];
            const float* wrow = &lds_in[oi * 256];
            #pragma unroll 8
            for (int co = 0; co < 256; co++)
                sum += wrow[co] * x_tile[co * 33 + pix];
            int p = row * WF + col0 + ph * 32 + pix;
            if (oi < 3) {
                scores[(size_t)n * NPROP + oi * (HF * WF) + p] = 1.0f / (1.0f + expf(-sum));
            } else {
                int ch = oi - 3;            // channel in [4A]; ch = ccomp*A + a
                int ccomp = ch / 3;
                int a = ch % 3;
                deltas[(((size_t)n * NPROP) + a * (HF * WF) + p) * 4 + ccomp] = sum;
            }
        }
    }
}

// ---------------------------------------------------------------------------
// K2: decode anchors + deltas -> clipped boxes, build 64-bit sort keys
// ---------------------------------------------------------------------------
__global__ __launch_bounds__(256) void decode_kernel(
    const float* __restrict__ scores, const float* __restrict__ deltas,
    float* __restrict__ boxes, unsigned long long* __restrict__ keys) {
    int t = blockIdx.x * 256 + threadIdx.x;   // NIMG * NPAD threads
    int n = t >> 16;
    int i = t & 0xFFFF;
    unsigned long long* kout = keys + ((size_t)n << 16);
    if (i >= NPROP) { kout[i] = 0ull; return; }

    // anchors indexed by (p, a) with p = i/3, a = i%3  (reference quirk)
    int pa = i / 3, aa = i - pa * 3;
    int ya = pa >> 7, xa = pa & 127;
    const float bs[3][4] = {{-91.f, -45.f, 91.f, 45.f},
                            {-64.f, -64.f, 64.f, 64.f},
                            {-45.f, -91.f, 45.f, 91.f}};
    float sxv = (float)(xa * 8), syv = (float)(ya * 8);
    float x1 = sxv + bs[aa][0], y1 = syv + bs[aa][1];
    float x2 = sxv + bs[aa][2], y2 = syv + bs[aa][3];
    float aw = x2 - x1, ah = y2 - y1;
    float acx = y1 + aw * 0.5f;   // (sic) matches reference swap
    float acy = x1 + ah * 0.5f;   // (sic)

    const float* d = deltas + (((size_t)n * NPROP) + i) * 4;
    float px = acx + d[0] * aw;
    float py = acy + d[1] * ah;
    float pw = aw * expf(d[2]);
    float ph = ah * expf(d[3]);
    float b0 = fminf(fmaxf(px - pw * 0.5f, 0.f), 1024.f);
    float b1 = fminf(fmaxf(py - ph * 0.5f, 0.f), 1024.f);
    float b2 = fminf(fmaxf(px + pw * 0.5f, 0.f), 1024.f);
    float b3 = fminf(fmaxf(py + ph * 0.5f, 0.f), 1024.f);

    float* bo = boxes + (((size_t)n * NPROP) + i) * 4;
    bo[0] = b0; bo[1] = b1; bo[2] = b2; bo[3] = b3;

    float s = scores[(size_t)n * NPROP + i];
    bool valid = (s > OBJ_THR) && ((b2 - b0) > 0.f) && ((b3 - b1) > 0.f);
    float ss = valid ? s : -__builtin_inff();
    unsigned u = __float_as_uint(ss);
    unsigned sv = (u & 0x80000000u) ? ~u : (u | 0x80000000u);
    kout[i] = ((unsigned long long)sv << 32) | (unsigned long long)(0xFFFFFFFFu - (unsigned)i);
}

// ---------------------------------------------------------------------------
// K3: one bitonic step (descending sort of 65536 keys per image)
// ---------------------------------------------------------------------------
__global__ __launch_bounds__(256) void bitonic_step_kernel(unsigned long long* __restrict__ keys,
                                                           int j, int k) {
    int t = blockIdx.x * 256 + threadIdx.x;   // NIMG * NPAD threads
    int n = t >> 16;
    int i = t & 0xFFFF;
    int ixj = i ^ j;
    if (ixj > i) {
        unsigned long long* K = keys + ((size_t)n << 16);
        unsigned long long a = K[i];
        unsigned long long b = K[ixj];
        bool up = ((i & k) == 0);
        bool sw = up ? (a < b) : (a > b);
        if (sw) { K[i] = b; K[ixj] = a; }
    }
}

// ---------------------------------------------------------------------------
// K4: greedy NMS on sorted top-2000 + compact top-300 output  (1 block/image)
// ---------------------------------------------------------------------------
__global__ __launch_bounds__(1024) void nms_kernel(const unsigned long long* __restrict__ keys,
                                                   const float* __restrict__ boxes,
                                                   float* __restrict__ out) {
    __shared__ float bx[PRE_NMS * 4];
    __shared__ float ar[PRE_NMS];
    __shared__ float sc[PRE_NMS];
    __shared__ int   kp[PRE_NMS];   // 0 rejected/invalid, 1 candidate, 2 kept
    __shared__ int   sup;

    int n = blockIdx.x;
    const unsigned long long* K = keys + ((size_t)n << 16);

    for (int i = threadIdx.x; i < PRE_NMS; i += blockDim.x) {
        unsigned long long key = K[i];
        unsigned sv  = (unsigned)(key >> 32);
        unsigned idx = 0xFFFFFFFFu - (unsigned)(key & 0xFFFFFFFFu);
        bool v = sv > 0x007FFFFFu;   // strictly above sortable(-inf)
        float b0 = 0.f, b1 = 0.f, b2 = 0.f, b3 = 0.f, s = 0.f;
        if (v) {
            const float* bp = boxes + (((size_t)n * NPROP) + idx) * 4;
            b0 = bp[0]; b1 = bp[1]; b2 = bp[2]; b3 = bp[3];
            unsigned u = (sv & 0x80000000u) ? (sv ^ 0x80000000u) : ~sv;
            s = __uint_as_float(u);
        }
        bx[i * 4 + 0] = b0; bx[i * 4 + 1] = b1;
        bx[i * 4 + 2] = b2; bx[i * 4 + 3] = b3;
        ar[i] = fmaxf(b2 - b0, 0.f) * fmaxf(b3 - b1, 0.f);
        sc[i] = s;
        kp[i] = v ? 1 : 0;
    }
    __syncthreads();

    for (int i = 0; i < PRE_NMS; i++) {
        if (threadIdx.x == 0) sup = 0;
        __syncthreads();
        float ib0 = bx[i * 4 + 0], ib1 = bx[i * 4 + 1];
        float ib2 = bx[i * 4 + 2], ib3 = bx[i * 4 + 3];
        float ia  = ar[i];
        int found = 0;
        for (int j = threadIdx.x; j < i; j += blockDim.x) {
            if (kp[j] == 2) {
                float jx1 = bx[j * 4 + 0], jy1 = bx[j * 4 + 1];
                float jx2 = bx[j * 4 + 2], jy2 = bx[j * 4 + 3];
                float ix1 = fmaxf(ib0, jx1), iy1 = fmaxf(ib1, jy1);
                float ix2 = fminf(ib2, jx2), iy2 = fminf(ib3, jy2);
                float inter = fmaxf(ix2 - ix1, 0.f) * fmaxf(iy2 - iy1, 0.f);
                float iou = inter / fmaxf(ia + ar[j] - inter, 1e-9f);
                if (iou > IOU_THR) found = 1;
            }
        }
        if (found) atomicOr(&sup, 1);
        __syncthreads();
        if (threadIdx.x == 0) kp[i] = (kp[i] == 1 && sup == 0) ? 2 : 0;
        __syncthreads();
    }

    for (int t = threadIdx.x; t < POST_NMS * 5; t += blockDim.x)
        out[(size_t)n * POST_NMS * 5 + t] = 0.f;
    __syncthreads();
    if (threadIdx.x == 0) {
        int cnt = 0;
        for (int i = 0; i < PRE_NMS && cnt < POST_NMS; i++) {
            if (kp[i] == 2) {
                float* o = out + (size_t)n * POST_NMS * 5 + cnt * 5;
                o[0] = bx[i * 4 + 0]; o[1] = bx[i * 4 + 1];
                o[2] = bx[i * 4 + 2]; o[3] = bx[i * 4 + 3];
                o[4] = sc[i];
                cnt++;
            }
        }
    }
}

// ---------------------------------------------------------------------------
extern "C" void kernel_launch(void* const* d_in, const int* in_sizes, int n_in,
                              void* d_out, int out_size, void* d_ws, size_t ws_size,
                              hipStream_t stream) {
    const float* feat   = (const float*)d_in[0];
    const float* conv_w = (const float*)d_in[1];
    const float* conv_b = (const float*)d_in[2];
    const float* det_w  = (const float*)d_in[3];
    const float* det_b  = (const float*)d_in[4];
    const float* reg_w  = (const float*)d_in[5];
    const float* reg_b  = (const float*)d_in[6];
    float* out = (float*)d_out;

    char* ws = (char*)d_ws;
    float* wt2                  = (float*)(ws);                       // 2,359,296 B
    float* scores               = (float*)(ws + 2359296);             //   786,432 B
    float* deltas               = (float*)(ws + 2359296 + 786432);    // 3,145,728 B
    float* boxes                = (float*)(ws + 2359296 + 786432 + 3145728);
    unsigned long long* keys    = (unsigned long long*)(ws + 2359296 + 786432 + 3145728 + 3145728);

    repack_w_kernel<<<2304, 256, 0, stream>>>(conv_w, wt2);
    conv_fused_kernel<<<NIMG * HF * (WF / TILE_W), 256, 0, stream>>>(
        feat, wt2, conv_b, det_w, det_b, reg_w, reg_b, scores, deltas);
    decode_kernel<<<(NIMG * NPAD) / 256, 256, 0, stream>>>(scores, deltas, boxes, keys);
    for (int k = 2; k <= NPAD; k <<= 1)
        for (int j = k >> 1; j > 0; j >>= 1)
            bitonic_step_kernel<<<(NIMG * NPAD) / 256, 256, 0, stream>>>(keys, j, k);
    nms_kernel<<<NIMG, 1024, 0, stream>>>(keys, boxes, out);
}